// TemporalLayer_18597208391971
// MI455X (gfx1250) — compile-verified
//
#include <hip/hip_runtime.h>
#include <hip/hip_bf16.h>

typedef float v2f __attribute__((ext_vector_type(2)));
typedef float v8f __attribute__((ext_vector_type(8)));
typedef unsigned int v4u __attribute__((ext_vector_type(4)));
typedef int v4i __attribute__((ext_vector_type(4)));
typedef int v8i __attribute__((ext_vector_type(8)));

#define T_DIM 8
#define N_DIM 4096
#define U_DIM 128
#define M_DIM 2048          // nodes per entity
#define NEG_SLOPE 0.01f
#define LN_EPS 1e-5f
#define NWAVES 4            // waves per attention workgroup (64 q rows)
#define PANEL_STRIDE 132    // padded LDS row stride (floats): 128 data + 4 pad

// ---- wave32 16-lane-half reductions (rows of a WMMA C tile live across lanes 0..15 / 16..31) ----
__device__ __forceinline__ float half16_max(float x) {
#pragma unroll
    for (int m = 8; m >= 1; m >>= 1) x = fmaxf(x, __shfl_xor(x, m, 32));
    return x;
}
__device__ __forceinline__ float half16_sum(float x) {
#pragma unroll
    for (int m = 8; m >= 1; m >>= 1) x += __shfl_xor(x, m, 32);
    return x;
}

__device__ __forceinline__ v8f wmma_f32(v2f a, v2f b, v8f c) {
    // D = A(16x4) * B(4x16) + C  (fp32 WMMA)
    return __builtin_amdgcn_wmma_f32_16x16x4_f32(false, a, false, b, (short)0, c, false, false);
}

// ---------------------------------------------------------------------------
// TDM: stage one 16x128 fp32 panel (global, row stride 128) into LDS with a
// padded row stride of 132 floats.  D# built per CDNA5 ISA.
//   pad_interval code 6 -> pad after every 2^(6+1)=128 DWORDs (one 512B row)
//   pad_amount   code 3 -> insert 4 DWORDs (16B) of padding
// ---------------------------------------------------------------------------
__device__ __forceinline__ void tdm_load_panel16x128(unsigned lds_byte_off,
                                                     const float* __restrict__ gsrc) {
    unsigned long long ga = (unsigned long long)(uintptr_t)gsrc;
    v4u g0;
    g0[0] = 1u;                                                  // count=1, user descriptor
    g0[1] = lds_byte_off;                                        // D#.lds_addr
    g0[2] = (unsigned)(ga & 0xFFFFFFFFu);                        // global_addr[31:0]
    g0[3] = (unsigned)((ga >> 32) & 0x01FFFFFFu) | (2u << 30);   // global_addr[56:32] | type=2
    v8i g1;
    g1[0] = (2 << 16) | (1 << 20) | (6 << 22) | (3 << 25);       // data_size=4B, pad_en, ivl=128dw, amt=4dw
    g1[1] = (128 << 16);                                         // tensor_dim0 = 128
    g1[2] = (16 << 16);                                          // tensor_dim1 = 16
    g1[3] = (128 << 16);                                         // tile_dim0   = 128
    g1[4] = 16;                                                  // tile_dim1   = 16
    g1[5] = 128;                                                 // tensor_dim0_stride = 128
    g1[6] = 0;
    g1[7] = 0;
    v4i gz = (v4i){0, 0, 0, 0};
#if __clang_major__ >= 23
    v8i gz8 = (v8i){0, 0, 0, 0, 0, 0, 0, 0};
    __builtin_amdgcn_tensor_load_to_lds(g0, g1, gz, gz, gz8, 0);
#else
    __builtin_amdgcn_tensor_load_to_lds(g0, g1, gz, gz, 0);
#endif
}

// ============================================================================
// Kernel 1: projections.  One wave computes a 16x128 output tile.
//   bid <  2048 : K[e][t] = S[e][t]   @ kw[e]
//   bid <  4096 : V[e][t] = S[e][t]   @ vw[e]
//   else        : Q[e]    = S[e][T-1] @ qw[e]
// ============================================================================
__global__ __launch_bounds__(32) void proj_kernel(
    const float* __restrict__ ns,
    const float* __restrict__ qw, const float* __restrict__ kw, const float* __restrict__ vw,
    float* __restrict__ wsQ, float* __restrict__ wsK, float* __restrict__ wsV)
{
    const int bid  = blockIdx.x;
    const int lane = threadIdx.x;
    const int c    = lane & 15;
    const int half = lane >> 4;

    const float* W; float* out; int e, t, tile;
    if (bid < 2048)      { int j = bid;        e = j >> 10; t = (j >> 7) & 7; tile = j & 127; W = kw + e*U_DIM*U_DIM; out = wsK + (((size_t)(e*T_DIM + t)*M_DIM) + tile*16)*U_DIM; }
    else if (bid < 4096) { int j = bid - 2048; e = j >> 10; t = (j >> 7) & 7; tile = j & 127; W = vw + e*U_DIM*U_DIM; out = wsV + (((size_t)(e*T_DIM + t)*M_DIM) + tile*16)*U_DIM; }
    else                 { int j = bid - 4096; e = j >> 7;  t = T_DIM - 1;   tile = j & 127; W = qw + e*U_DIM*U_DIM; out = wsQ + ((size_t)(e*M_DIM) + tile*16)*U_DIM; }

    const float* S = ns + (((size_t)t*N_DIM) + e*M_DIM + tile*16)*U_DIM;

    v8f acc[8];
#pragma unroll
    for (int nt = 0; nt < 8; ++nt) acc[nt] = (v8f){0,0,0,0,0,0,0,0};

#pragma unroll 4
    for (int ki = 0; ki < 32; ++ki) {
        const int kk = 4*ki;
        v2f a = *(const v2f*)(S + (size_t)c*U_DIM + kk + 2*half);
#pragma unroll
        for (int nt = 0; nt < 8; ++nt) {
            v2f b;
            b.x = W[(size_t)(kk + 2*half)    *U_DIM + nt*16 + c];
            b.y = W[(size_t)(kk + 2*half + 1)*U_DIM + nt*16 + c];
            acc[nt] = wmma_f32(a, b, acc[nt]);
        }
    }

#pragma unroll
    for (int nt = 0; nt < 8; ++nt)
#pragma unroll
        for (int r = 0; r < 8; ++r)
            out[(size_t)(r + 8*half)*U_DIM + nt*16 + c] = acc[nt][r];
}

// ============================================================================
// Kernel 2: fused flash attention + LayerNorm + FC + leaky-relu + residual.
//   4 waves per block; each wave owns 16 q rows.  K/V panels (16x128) are
//   double-buffered in LDS via TDM tensor_load_to_lds and shared by all waves.
//   grid = 2 entities * 8 timesteps * 32 row blocks = 512 blocks of 128.
// ============================================================================
__global__ __launch_bounds__(32*NWAVES) void attn_kernel(
    const float* __restrict__ fcw,
    const float* __restrict__ lnw, const float* __restrict__ lnb,
    const int*   __restrict__ acc_idx, const int* __restrict__ stock_idx,
    const float* __restrict__ wsQ, const float* __restrict__ wsK, const float* __restrict__ wsV,
    float* __restrict__ out)
{
    __shared__ float ldsK[2][16 * PANEL_STRIDE];   // double-buffered K panel (TDM dest)
    __shared__ float ldsV[2][16 * PANEL_STRIDE];   // double-buffered V panel (TDM dest)
    __shared__ float ldsP[NWAVES][16 * 17];        // per-wave P tile bounce
    __shared__ float ldsY[NWAVES][16 * PANEL_STRIDE]; // per-wave LN-output bounce

    const int bid  = blockIdx.x;
    const int e    = bid >> 8;
    const int t    = (bid >> 5) & 7;
    const int rb   = bid & 31;

    const int w    = threadIdx.x >> 5;
    const int lane = threadIdx.x & 31;
    const int c    = lane & 15;
    const int half = lane >> 4;
    const int m0   = rb*64 + w*16;                 // this wave's 16 q rows

    const float* Qp = wsQ + ((size_t)(e*M_DIM) + m0)*U_DIM;
    const float* Kb = wsK + ((size_t)(e*T_DIM + t)*M_DIM)*U_DIM;
    const float* Vb = wsV + ((size_t)(e*T_DIM + t)*M_DIM)*U_DIM;

    // Prologue: wave 0 kicks off TDM for panel 0 into buffer 0.
    if (w == 0) {
        tdm_load_panel16x128((unsigned)(uintptr_t)&ldsK[0][0], Kb);
        tdm_load_panel16x128((unsigned)(uintptr_t)&ldsV[0][0], Vb);
    }

    // Preload all 32 A-fragments of this wave's 16x128 q block (reused every panel).
    v2f qf[32];
#pragma unroll
    for (int i = 0; i < 32; ++i)
        qf[i] = *(const v2f*)(Qp + (size_t)c*U_DIM + 4*i + 2*half);

    v8f o[8];
#pragma unroll
    for (int vt = 0; vt < 8; ++vt) o[vt] = (v8f){0,0,0,0,0,0,0,0};
    float runmax[8], runsum[8];
#pragma unroll
    for (int r = 0; r < 8; ++r) { runmax[r] = -INFINITY; runsum[r] = 0.f; }

    const int NB = M_DIM / 16;
#pragma unroll 1
    for (int mb = 0; mb < NB; ++mb) {
        const int buf = mb & 1;

        // Prefetch next panel into the other buffer, then retire the current
        // batch (TENSORcnt is in-order per wave: <=2 leaves only the new pair).
        if (w == 0) {
            if (mb + 1 < NB) {
                tdm_load_panel16x128((unsigned)(uintptr_t)&ldsK[buf ^ 1][0], Kb + (size_t)(mb + 1)*16*U_DIM);
                tdm_load_panel16x128((unsigned)(uintptr_t)&ldsV[buf ^ 1][0], Vb + (size_t)(mb + 1)*16*U_DIM);
                __builtin_amdgcn_s_wait_tensorcnt(2);
            } else {
                __builtin_amdgcn_s_wait_tensorcnt(0);
            }
        }
        __syncthreads();   // panel mb visible to all waves

        const float* Kp = ldsK[buf];
        const float* Vp = ldsV[buf];

        // score tile = q(16x128) @ K_panel(16x128)^T  -> 16x16 (B frags from LDS)
        v8f s = (v8f){0,0,0,0,0,0,0,0};
#pragma unroll
        for (int i = 0; i < 32; ++i) {
            v2f b = *(const v2f*)(Kp + c*PANEL_STRIDE + 4*i + 2*half);
            s = wmma_f32(qf[i], b, s);
        }

        // online softmax update
        float alpha[8];
#pragma unroll
        for (int r = 0; r < 8; ++r) {
            float bmax   = half16_max(s[r]);
            float newmax = fmaxf(runmax[r], bmax);
            alpha[r]     = __expf(runmax[r] - newmax);
            float p      = __expf(s[r] - newmax);
            runsum[r]    = runsum[r]*alpha[r] + half16_sum(p);
            runmax[r]    = newmax;
            ldsP[w][(r + 8*half)*17 + c] = p;
        }
#pragma unroll
        for (int vt = 0; vt < 8; ++vt)
#pragma unroll
            for (int r = 0; r < 8; ++r) o[vt][r] *= alpha[r];

        // O += P(16x16) @ V_panel(16x128)
#pragma unroll
        for (int kki = 0; kki < 4; ++kki) {
            const int kk = 4*kki;
            v2f a;
            a.x = ldsP[w][c*17 + kk + 2*half];
            a.y = ldsP[w][c*17 + kk + 2*half + 1];
#pragma unroll
            for (int vt = 0; vt < 8; ++vt) {
                v2f b;
                b.x = Vp[(kk + 2*half)    *PANEL_STRIDE + vt*16 + c];
                b.y = Vp[(kk + 2*half + 1)*PANEL_STRIDE + vt*16 + c];
                o[vt] = wmma_f32(a, b, o[vt]);
            }
        }
        __syncthreads();   // all waves done with buf before next prefetch overwrites it
    }

    // finish softmax normalization
#pragma unroll
    for (int r = 0; r < 8; ++r) {
        float inv = 1.f / runsum[r];
#pragma unroll
        for (int vt = 0; vt < 8; ++vt) o[vt][r] *= inv;
    }

    // LayerNorm over the 128-wide rows; write in A-consumable LDS layout
    float lw[8], lb[8];
#pragma unroll
    for (int vt = 0; vt < 8; ++vt) { lw[vt] = lnw[vt*16 + c]; lb[vt] = lnb[vt*16 + c]; }
#pragma unroll
    for (int r = 0; r < 8; ++r) {
        float part = 0.f;
#pragma unroll
        for (int vt = 0; vt < 8; ++vt) part += o[vt][r];
        float mu = half16_sum(part) * (1.f/128.f);
        float vp = 0.f;
#pragma unroll
        for (int vt = 0; vt < 8; ++vt) { float d = o[vt][r] - mu; vp += d*d; }
        float rstd = rsqrtf(half16_sum(vp) * (1.f/128.f) + LN_EPS);
#pragma unroll
        for (int vt = 0; vt < 8; ++vt)
            ldsY[w][(r + 8*half)*PANEL_STRIDE + vt*16 + c] = (o[vt][r] - mu)*rstd*lw[vt] + lb[vt];
    }

    // FC: Y(16x128) @ fcw[e](128x128)
    const float* W = fcw + (size_t)e*U_DIM*U_DIM;
    v8f f[8];
#pragma unroll
    for (int nt = 0; nt < 8; ++nt) f[nt] = (v8f){0,0,0,0,0,0,0,0};
#pragma unroll 4
    for (int ki = 0; ki < 32; ++ki) {
        const int kk = 4*ki;
        v2f a;
        a.x = ldsY[w][c*PANEL_STRIDE + kk + 2*half];
        a.y = ldsY[w][c*PANEL_STRIDE + kk + 2*half + 1];
#pragma unroll
        for (int nt = 0; nt < 8; ++nt) {
            v2f b;
            b.x = W[(size_t)(kk + 2*half)    *U_DIM + nt*16 + c];
            b.y = W[(size_t)(kk + 2*half + 1)*U_DIM + nt*16 + c];
            f[nt] = wmma_f32(a, b, f[nt]);
        }
    }

    // leaky-relu + residual(q) + scatter via entity index arrays
    const int* idx = (e == 0) ? acc_idx : stock_idx;
    int node[8];
#pragma unroll
    for (int r = 0; r < 8; ++r) node[r] = idx[m0 + r + 8*half];
#pragma unroll
    for (int nt = 0; nt < 8; ++nt)
#pragma unroll
        for (int r = 0; r < 8; ++r) {
            float x = f[nt][r];
            x = (x > 0.f) ? x : NEG_SLOPE * x;
            x += Qp[(size_t)(r + 8*half)*U_DIM + nt*16 + c];
            out[((size_t)t*N_DIM + node[r])*U_DIM + nt*16 + c] = x;
        }
}

// ============================================================================
extern "C" void kernel_launch(void* const* d_in, const int* in_sizes, int n_in,
                              void* d_out, int out_size, void* d_ws, size_t ws_size,
                              hipStream_t stream) {
    const float* ns   = (const float*)d_in[0];
    const float* qw   = (const float*)d_in[1];
    const float* kw   = (const float*)d_in[2];
    const float* vw   = (const float*)d_in[3];
    const float* fcw  = (const float*)d_in[4];
    const float* lnw  = (const float*)d_in[5];
    const float* lnb  = (const float*)d_in[6];
    const int*   ai   = (const int*)d_in[7];
    const int*   si   = (const int*)d_in[8];
    float*       outp = (float*)d_out;

    float* wsf = (float*)d_ws;
    float* wsQ = wsf;                                   // 2 * 2048 * 128
    float* wsK = wsQ + (size_t)2*M_DIM*U_DIM;           // 2 * 8 * 2048 * 128
    float* wsV = wsK + (size_t)2*T_DIM*M_DIM*U_DIM;     // 2 * 8 * 2048 * 128

    proj_kernel<<<dim3(4096 + 256), dim3(32),        0, stream>>>(ns, qw, kw, vw, wsQ, wsK, wsV);
    attn_kernel<<<dim3(512),        dim3(32*NWAVES), 0, stream>>>(fcw, lnw, lnb, ai, si, wsQ, wsK, wsV, outp);
}